// MultiEmbodimentActionEncoder_25795573580026
// MI455X (gfx1250) — compile-verified
//
#include <hip/hip_runtime.h>
#include <hip/hip_bf16.h>

// MultiEmbodimentActionEncoder for MI455X (gfx1250), f32 WMMA path, v2.
//
// Shapes: B=128, H=16, ACTION_DIM=32, HIDDEN=256, EMBED=512, NUM_CAT=16.
// One workgroup (8 wave32) per batch row (one M=16 tile, single category).
// Each wave computes all of its N-tiles in a single K-pass: one ds_load_b64
// of the A operand feeds 2-4 independent v_wmma_f32_16x16x4_f32 chains, and
// sibling-tile B loads share one VGPR address with immediate offsets.

#define ACTION_DIM 32
#define EMBED_DIM  512
#define HIDDEN_DIM 256
#define HORIZON    16

typedef __attribute__((ext_vector_type(2))) float v2f;
typedef __attribute__((ext_vector_type(8))) float v8f;

__device__ __forceinline__ v8f wmma_f32(v2f a, v2f b, v8f c) {
  // 8 args: (neg_a, A, neg_b, B, c_mod, C, reuse_a, reuse_b)
  return __builtin_amdgcn_wmma_f32_16x16x4_f32(
      false, a, false, b, (short)0, c, false, false);
}

__global__ __launch_bounds__(256)
void MultiEmbodimentActionEncoder_kernel(
    const float* __restrict__ X,    // [B,16,32]
    const int*   __restrict__ cids, // [B]
    const float* __restrict__ W1,   // [16,32,256]
    const float* __restrict__ b1,   // [16,256]
    const float* __restrict__ W2,   // [16,256,256]
    const float* __restrict__ b2,   // [16,256]
    const float* __restrict__ W3,   // [16,256,512]
    const float* __restrict__ b3,   // [16,512]
    float*       __restrict__ out)  // [B,16,512]
{
  __shared__ float Xs[HORIZON * ACTION_DIM];   //  2 KB
  __shared__ float H1[HORIZON * HIDDEN_DIM];   // 16 KB
  __shared__ float H2[HORIZON * HIDDEN_DIM];   // 16 KB

  const int bid  = blockIdx.x;
  const int tid  = threadIdx.x;
  const int wave = tid >> 5;
  const int lane = tid & 31;
  const int m16  = lane & 15;  // row (A) / col (B,C,D) index within tile
  const int kh   = lane >> 4;  // K half-select for A/B, M half-select for C/D

  const int cid = cids[bid];   // uniform per block -> scalar load
  const float* __restrict__ W1c = W1 + (size_t)cid * ACTION_DIM * HIDDEN_DIM;
  const float* __restrict__ b1c = b1 + (size_t)cid * HIDDEN_DIM;
  const float* __restrict__ W2c = W2 + (size_t)cid * HIDDEN_DIM * HIDDEN_DIM;
  const float* __restrict__ b2c = b2 + (size_t)cid * HIDDEN_DIM;
  const float* __restrict__ W3c = W3 + (size_t)cid * HIDDEN_DIM * EMBED_DIM;
  const float* __restrict__ b3c = b3 + (size_t)cid * EMBED_DIM;
  const float* __restrict__ Xg  = X  + (size_t)bid * HORIZON * ACTION_DIM;

  // Stage this row's activations: 512 floats, 256 threads x float2 (coalesced).
  {
    float2 v = ((const float2*)Xg)[tid];
    ((float2*)Xs)[tid] = v;
  }
  __syncthreads();

  const int n0 = wave * 16 + m16;   // this wave's first output column

  // ---------------- Layer 1: [16x32] @ W1[cid] (32x256), +b1, ReLU, +PE ----
  // Wave handles columns n0 and n0+128 in one K-pass.
  {
    const float* __restrict__ bp = W1c + (size_t)(2 * kh) * HIDDEN_DIM + n0;
    v8f acc0 = {}, acc1 = {};
#pragma unroll
    for (int kb = 0; kb < ACTION_DIM / 4; ++kb) {
      v2f a = *(const v2f*)&Xs[m16 * ACTION_DIM + kb * 4 + 2 * kh];
      v2f bA, bB;
      bA.x = bp[kb * 4 * HIDDEN_DIM];
      bA.y = bp[kb * 4 * HIDDEN_DIM + HIDDEN_DIM];
      bB.x = bp[kb * 4 * HIDDEN_DIM + 128];
      bB.y = bp[kb * 4 * HIDDEN_DIM + HIDDEN_DIM + 128];
      acc0 = wmma_f32(a, bA, acc0);
      acc1 = wmma_f32(a, bB, acc1);
    }
    const float coef = -9.210340371976184f / (float)HIDDEN_DIM; // -ln(1e4)/256
#pragma unroll
    for (int half = 0; half < 2; ++half) {
      const int   n    = n0 + half * 128;
      const v8f&  acc  = half ? acc1 : acc0;
      const float bias = b1c[n];
      const float dt   = __expf((float)(n & ~1) * coef);
#pragma unroll
      for (int r = 0; r < 8; ++r) {
        const int m = r + 8 * kh;
        float v = fmaxf(acc[r] + bias, 0.0f);
        const float ang = (float)m * dt;
        const float s = __sinf(ang);
        const float c = __cosf(ang);
        v += (n & 1) ? c : s;               // v_cndmask, no exec branching
        H1[m * HIDDEN_DIM + n] = v;
      }
    }
  }
  __syncthreads();

  // ---------------- Layer 2: [16x256] @ W2[cid] (256x256), +b2, ReLU -------
  {
    const float* __restrict__ bp = W2c + (size_t)(2 * kh) * HIDDEN_DIM + n0;
    const float* __restrict__ ap = &H1[m16 * HIDDEN_DIM + 2 * kh];
    v8f acc0 = {}, acc1 = {};
#pragma unroll 8
    for (int kb = 0; kb < HIDDEN_DIM / 4; ++kb) {
      v2f a = *(const v2f*)&ap[kb * 4];
      v2f bA, bB;
      bA.x = bp[kb * 4 * HIDDEN_DIM];
      bA.y = bp[kb * 4 * HIDDEN_DIM + HIDDEN_DIM];
      bB.x = bp[kb * 4 * HIDDEN_DIM + 128];
      bB.y = bp[kb * 4 * HIDDEN_DIM + HIDDEN_DIM + 128];
      acc0 = wmma_f32(a, bA, acc0);
      acc1 = wmma_f32(a, bB, acc1);
    }
#pragma unroll
    for (int half = 0; half < 2; ++half) {
      const int   n    = n0 + half * 128;
      const v8f&  acc  = half ? acc1 : acc0;
      const float bias = b2c[n];
#pragma unroll
      for (int r = 0; r < 8; ++r) {
        const int m = r + 8 * kh;
        H2[m * HIDDEN_DIM + n] = fmaxf(acc[r] + bias, 0.0f);
      }
    }
  }
  __syncthreads();

  // ---------------- Layer 3: [16x256] @ W3[cid] (256x512), +b3 -------------
  // Wave handles columns n0 + {0,128,256,384} in one K-pass (4 acc chains).
  {
    float* __restrict__ outg = out + (size_t)bid * HORIZON * EMBED_DIM;
    const float* __restrict__ bp = W3c + (size_t)(2 * kh) * EMBED_DIM + n0;
    const float* __restrict__ ap = &H2[m16 * HIDDEN_DIM + 2 * kh];
    v8f acc0 = {}, acc1 = {}, acc2 = {}, acc3 = {};
#pragma unroll 8
    for (int kb = 0; kb < HIDDEN_DIM / 4; ++kb) {
      v2f a = *(const v2f*)&ap[kb * 4];
      v2f bA, bB, bC, bD;
      bA.x = bp[kb * 4 * EMBED_DIM];
      bA.y = bp[kb * 4 * EMBED_DIM + EMBED_DIM];
      bB.x = bp[kb * 4 * EMBED_DIM + 128];
      bB.y = bp[kb * 4 * EMBED_DIM + EMBED_DIM + 128];
      bC.x = bp[kb * 4 * EMBED_DIM + 256];
      bC.y = bp[kb * 4 * EMBED_DIM + EMBED_DIM + 256];
      bD.x = bp[kb * 4 * EMBED_DIM + 384];
      bD.y = bp[kb * 4 * EMBED_DIM + EMBED_DIM + 384];
      acc0 = wmma_f32(a, bA, acc0);
      acc1 = wmma_f32(a, bB, acc1);
      acc2 = wmma_f32(a, bC, acc2);
      acc3 = wmma_f32(a, bD, acc3);
    }
#pragma unroll
    for (int q = 0; q < 4; ++q) {
      const int   n    = n0 + q * 128;
      const v8f&  acc  = (q == 0) ? acc0 : (q == 1) ? acc1 : (q == 2) ? acc2 : acc3;
      const float bias = b3c[n];
#pragma unroll
      for (int r = 0; r < 8; ++r) {
        const int m = r + 8 * kh;
        outg[(size_t)m * EMBED_DIM + n] = acc[r] + bias;
      }
    }
  }
}

extern "C" void kernel_launch(void* const* d_in, const int* in_sizes, int n_in,
                              void* d_out, int out_size, void* d_ws, size_t ws_size,
                              hipStream_t stream) {
  const float* X    = (const float*)d_in[0];
  const int*   cids = (const int*)  d_in[1];
  const float* W1   = (const float*)d_in[2];
  const float* b1   = (const float*)d_in[3];
  const float* W2   = (const float*)d_in[4];
  const float* b2   = (const float*)d_in[5];
  const float* W3   = (const float*)d_in[6];
  const float* b3   = (const float*)d_in[7];
  float* out = (float*)d_out;

  const int Bn = in_sizes[1];  // one workgroup per batch row
  MultiEmbodimentActionEncoder_kernel<<<Bn, 256, 0, stream>>>(
      X, cids, W1, b1, W2, b2, W3, b3, out);
}